// QuantumEmbedding_50964081935097
// MI455X (gfx1250) — compile-verified
//
#include <hip/hip_runtime.h>
#include <hip/hip_bf16.h>

typedef __attribute__((ext_vector_type(2))) float v2f;
typedef __attribute__((ext_vector_type(8))) float v8f;

#define BATCH_N        32768
#define JDIM           2048
#define ROWS_PER_BLOCK 32      // 2 row-tiles of 16
#define WAVES_PER_TILE 4       // K split 4 ways
#define KSLICE         (JDIM / WAVES_PER_TILE)   // 512
#define NTHREADS       256     // 8 waves (wave32)

__global__ __launch_bounds__(NTHREADS)
void quantum_embed_wmma_f32(const float* __restrict__ x,
                            const float* __restrict__ W,       // [J][2]
                            const float* __restrict__ phases,  // [J]
                            float* __restrict__ out)           // [B][2]
{
    __shared__ float s_ph[JDIM];        // phases, 8 KB
    __shared__ float s_wt[2][JDIM];     // W transposed: s_wt[e][j], 16 KB
    __shared__ float s_red[8][32];      // per-wave partial D tiles, 1 KB

    const int tid = threadIdx.x;

    // Stage phases and transposed weights into LDS (keeps VMEM path clean
    // for streaming x at full HBM bandwidth).
    for (int j = tid; j < JDIM; j += NTHREADS) {
        s_ph[j] = phases[j];
        float2 w = ((const float2*)W)[j];   // W[j][0], W[j][1]
        s_wt[0][j] = w.x;
        s_wt[1][j] = w.y;
    }
    __syncthreads();

    const int wave = tid >> 5;
    const int lane = tid & 31;
    const int tile = wave >> 2;          // 0..1 : which 16-row tile
    const int kslc = wave & 3;           // 0..3 : which K slice
    const int m    = lane & 15;          // A: M index / B,C,D: N index
    const int h    = lane >> 4;          // lane half selects K pair
    const bool nvalid = (m < 2);         // only N<2 columns are real output
    const int  nc     = nvalid ? m : 0;  // safe LDS row for B loads

    const int row = blockIdx.x * ROWS_PER_BLOCK + tile * 16 + m;
    const float* __restrict__ xrow = x + (size_t)row * JDIM;
    const int kbase = kslc * KSLICE;

    v8f c = {};  // f32 accumulator tile (16x16, 8 VGPRs)

    // 128 WMMA steps of K=4 each. Lane L covers K = {2h, 2h+1} of each step.
    #pragma unroll 4
    for (int it = 0; it < KSLICE / 4; ++it) {
        const int c0 = kbase + it * 4 + 2 * h;

        float2 xv = *(const float2*)(xrow + c0);       // global stream of x
        float2 pv = *(const float2*)(&s_ph[c0]);       // LDS
        float2 wv = *(const float2*)(&s_wt[nc][c0]);   // LDS (B operand)

        v2f a, b;
        a.x = __cosf(xv.x * pv.x);                     // v_cos_f32 path
        a.y = __cosf(xv.y * pv.y);
        b.x = nvalid ? wv.x : 0.0f;                    // VALU select, EXEC stays full
        b.y = nvalid ? wv.y : 0.0f;

        // D = A(16x4 f32) x B(4x16 f32) + C, full f32 precision
        c = __builtin_amdgcn_wmma_f32_16x16x4_f32(
                /*neg_a=*/false, a, /*neg_b=*/false, b,
                /*c_mod=*/(short)0, c, /*reuse_a=*/false, /*reuse_b=*/false);
    }

    // Spill the useful columns (N=0,1 -> lanes 0,1,16,17) of this wave's
    // partial D tile. VGPR v holds M = v + 8*h.
    if (nvalid) {
        #pragma unroll
        for (int v = 0; v < 8; ++v) {
            int mrow = v + 8 * h;
            s_red[wave][mrow * 2 + m] = c[v];
        }
    }
    __syncthreads();

    // Deterministic cross-K-slice reduction + contiguous store.
    if (tid < 2 * 32) {
        int t   = tid >> 5;     // row-tile
        int idx = tid & 31;     // m*2 + e within tile
        float acc = s_red[t * 4 + 0][idx] + s_red[t * 4 + 1][idx]
                  + s_red[t * 4 + 2][idx] + s_red[t * 4 + 3][idx];
        out[blockIdx.x * (ROWS_PER_BLOCK * 2) + tid] = acc;
    }
}

extern "C" void kernel_launch(void* const* d_in, const int* in_sizes, int n_in,
                              void* d_out, int out_size, void* d_ws, size_t ws_size,
                              hipStream_t stream) {
    const float* x      = (const float*)d_in[0];   // [32768, 2048] f32
    const float* W      = (const float*)d_in[1];   // [2048, 2]     f32
    const float* phases = (const float*)d_in[2];   // [2048, 1]     f32
    float* out          = (float*)d_out;           // [32768, 2]    f32

    dim3 grid(BATCH_N / ROWS_PER_BLOCK);           // 1024 blocks
    dim3 block(NTHREADS);                          // 8 waves/block
    quantum_embed_wmma_f32<<<grid, block, 0, stream>>>(x, W, phases, out);
}